// DHSpace_22247930593796
// MI455X (gfx1250) — compile-verified
//
#include <hip/hip_runtime.h>
#include <hip/hip_bf16.h>

// ---------------- problem constants ----------------
constexpr int NN    = 50000;   // nodes
constexpr int EE    = 600000;  // edges
constexpr int HID   = 128;
constexpr int HEADS = 8;
constexpr int DK    = 16;
constexpr int K_N   = 2;
constexpr int K_R   = 4;

typedef __attribute__((ext_vector_type(2))) float v2f;
typedef __attribute__((ext_vector_type(8))) float v8f;

// ---------------- helpers ----------------
__device__ inline unsigned f2ord(float f) {
    unsigned u = __float_as_uint(f);
    return (u & 0x80000000u) ? ~u : (u | 0x80000000u);
}
__device__ inline float ord2f(unsigned u) {
    return (u & 0x80000000u) ? __uint_as_float(u & 0x7fffffffu)
                             : __uint_as_float(~u);
}
// ordered-uint encoding of -inf: bits(-inf)=0xFF800000 (negative) -> ~bits
constexpr unsigned ORD_NEG_INF = 0x007FFFFFu;

// ---------------------------------------------------------------------------
// Typed linear: out[n, n0:n0+16] = in[n,:] @ W[type[n]] (+ bias) (+ residual)
// One wave per 16(M-nodes) x 16(N-out) tile.  K=128 in steps of 4 using
// V_WMMA_F32_16X16X4_F32; two accumulator chains (one per node type), select
// per output row afterwards.
// A frag (16x4 f32): lanes 0-15 row M=lane hold K=kk,kk+1; lanes 16-31 row
// M=lane-16 hold K=kk+2,kk+3.  B frag mirrors A (K indexed by lane-half).
// ---------------------------------------------------------------------------
__global__ __launch_bounds__(32) void typed_linear16(
    const float* __restrict__ in, const float* __restrict__ W,
    const float* __restrict__ bias, const int* __restrict__ node_type,
    const float* __restrict__ resid, float* __restrict__ out)
{
    const int m0   = blockIdx.x * 16;
    const int n0   = blockIdx.y * 16;
    const int lane = threadIdx.x;
    const int half = lane >> 4;       // 0: lanes 0-15, 1: lanes 16-31
    const int lcol = lane & 15;

    const float* a_base = in + (size_t)(m0 + lcol) * HID + half * 2;
    const float* w0 = W + 0 * HID * HID + n0 + lcol;
    const float* w1 = W + 1 * HID * HID + n0 + lcol;

    v8f acc0 = {}; v8f acc1 = {};
    #pragma unroll 4
    for (int kk = 0; kk < HID; kk += 4) {
        v2f a; a.x = a_base[kk]; a.y = a_base[kk + 1];
        const int krow = kk + half * 2;
        v2f b0; b0.x = w0[(size_t)krow * HID]; b0.y = w0[(size_t)(krow + 1) * HID];
        v2f b1; b1.x = w1[(size_t)krow * HID]; b1.y = w1[(size_t)(krow + 1) * HID];
        acc0 = __builtin_amdgcn_wmma_f32_16x16x4_f32(false, a, false, b0,
                                                     (short)0, acc0, false, false);
        acc1 = __builtin_amdgcn_wmma_f32_16x16x4_f32(false, a, false, b1,
                                                     (short)0, acc1, false, false);
    }

    // D layout: VGPR i -> (M=i, N=lane) for lanes 0-15; (M=i+8, N=lane-16)
    #pragma unroll
    for (int i = 0; i < 8; ++i) {
        const int row = m0 + (half ? i + 8 : i);
        const int t   = node_type[row];
        float val = t ? acc1[i] : acc0[i];
        val += bias[t * HID + n0 + lcol];
        if (resid) val += resid[(size_t)row * HID + n0 + lcol];
        out[(size_t)row * HID + n0 + lcol] = val;
    }
}

// ---------------------------------------------------------------------------
// Per-(relation,head) 16x16 transforms: k_rel[r,n,h,:] = k[n,h,:] @ rel_att[r,h]
// and v_rel[r,n,h,:] = v[n,h,:] @ rel_msg[r,h].  One wave per (16-node tile,
// r*HEADS+h); K=16 in 4 steps of V_WMMA_F32_16X16X4_F32 (2 chains).
// ---------------------------------------------------------------------------
__global__ __launch_bounds__(32) void rel_transform16(
    const float* __restrict__ kmat, const float* __restrict__ vmat,
    const float* __restrict__ rel_att, const float* __restrict__ rel_msg,
    float* __restrict__ k_rel, float* __restrict__ v_rel)
{
    const int m0   = blockIdx.x * 16;
    const int rh   = blockIdx.y;          // r*HEADS + h
    const int h    = rh & (HEADS - 1);
    const int r    = rh >> 3;
    const int lane = threadIdx.x;
    const int half = lane >> 4;
    const int lcol = lane & 15;

    const float* ka = kmat + (size_t)(m0 + lcol) * HID + h * DK + half * 2;
    const float* va = vmat + (size_t)(m0 + lcol) * HID + h * DK + half * 2;
    const float* Ba = rel_att + (size_t)rh * DK * DK + lcol;   // [16][16] row-major
    const float* Bm = rel_msg + (size_t)rh * DK * DK + lcol;

    v8f acck = {}; v8f accv = {};
    #pragma unroll
    for (int kk = 0; kk < DK; kk += 4) {
        const int krow = kk + half * 2;
        v2f a1; a1.x = ka[kk]; a1.y = ka[kk + 1];
        v2f a2; a2.x = va[kk]; a2.y = va[kk + 1];
        v2f b1; b1.x = Ba[krow * DK]; b1.y = Ba[(krow + 1) * DK];
        v2f b2; b2.x = Bm[krow * DK]; b2.y = Bm[(krow + 1) * DK];
        acck = __builtin_amdgcn_wmma_f32_16x16x4_f32(false, a1, false, b1,
                                                     (short)0, acck, false, false);
        accv = __builtin_amdgcn_wmma_f32_16x16x4_f32(false, a2, false, b2,
                                                     (short)0, accv, false, false);
    }

    #pragma unroll
    for (int i = 0; i < 8; ++i) {
        const int row = m0 + (half ? i + 8 : i);
        const size_t off = (((size_t)r * NN + row) * HEADS + h) * DK + lcol;
        k_rel[off] = acck[i];
        v_rel[off] = accv[i];
    }
}

// ---------------------------------------------------------------------------
// Init: segment-max keys to ordered(-inf), denom/aggr to zero.
// ---------------------------------------------------------------------------
__global__ void init_kernel(unsigned* __restrict__ mord,
                            float* __restrict__ denom,
                            float* __restrict__ aggr)
{
    const int idx = blockIdx.x * blockDim.x + threadIdx.x;
    if (idx < NN * HEADS) { mord[idx] = ORD_NEG_INF; denom[idx] = 0.0f; }
    if (idx < NN * HID)   { aggr[idx] = 0.0f; }
}

// ---------------------------------------------------------------------------
// Edge scores: att[e,h] = <q[tgt,h,:], k_rel[rel,src,h,:]> * pri[rel,h] / 4
// plus segment-max via ordered-uint atomicMax.  Thread per (edge, head).
// ---------------------------------------------------------------------------
__global__ void score_kernel(const float* __restrict__ q,
                             const float* __restrict__ k_rel,
                             const int* __restrict__ ei,
                             const int* __restrict__ erel,
                             const float* __restrict__ rel_pri,
                             float* __restrict__ att,
                             unsigned* __restrict__ mord)
{
    const int idx = blockIdx.x * blockDim.x + threadIdx.x;
    const int e = idx >> 3, h = idx & 7;
    if (e >= EE) return;
    const int s = ei[e], t = ei[EE + e], r = erel[e];
    const float4* qp = (const float4*)(q + (size_t)t * HID + h * DK);
    const float4* kp = (const float4*)(k_rel + (((size_t)r * NN + s) * HEADS + h) * DK);
    float dot = 0.0f;
    #pragma unroll
    for (int j = 0; j < 4; ++j) {
        float4 a = qp[j], b = kp[j];
        dot += a.x * b.x + a.y * b.y + a.z * b.z + a.w * b.w;
    }
    const float val = dot * rel_pri[r * HEADS + h] * 0.25f;  // 1/sqrt(16)
    att[(size_t)e * HEADS + h] = val;
    atomicMax(mord + (size_t)t * HEADS + h, f2ord(val));
}

// ---------------------------------------------------------------------------
// exp(att - max) and segment-sum of denominators.
// ---------------------------------------------------------------------------
__global__ void exp_kernel(const int* __restrict__ ei,
                           const unsigned* __restrict__ mord,
                           float* __restrict__ att,
                           float* __restrict__ denom)
{
    const int idx = blockIdx.x * blockDim.x + threadIdx.x;
    const int e = idx >> 3, h = idx & 7;
    if (e >= EE) return;
    const int t = ei[EE + e];
    const float m = ord2f(mord[(size_t)t * HEADS + h]);
    const float v = __expf(att[(size_t)e * HEADS + h] - m);
    att[(size_t)e * HEADS + h] = v;
    atomicAdd(denom + (size_t)t * HEADS + h, v);
}

// ---------------------------------------------------------------------------
// Normalized scatter-add aggregation: aggr[tgt, h*16+d] += w * v_rel[rel,src,h,d]
// ---------------------------------------------------------------------------
__global__ void aggr_kernel(const int* __restrict__ ei,
                            const int* __restrict__ erel,
                            const float* __restrict__ att,
                            const float* __restrict__ denom,
                            const float* __restrict__ v_rel,
                            float* __restrict__ aggr)
{
    const int idx = blockIdx.x * blockDim.x + threadIdx.x;
    const int e = idx >> 3, h = idx & 7;
    if (e >= EE) return;
    const int s = ei[e], t = ei[EE + e], r = erel[e];
    const float w = att[(size_t)e * HEADS + h] /
                    fmaxf(denom[(size_t)t * HEADS + h], 1e-9f);
    const float* mp = v_rel + (((size_t)r * NN + s) * HEADS + h) * DK;
    float* op = aggr + (size_t)t * HID + h * DK;
    #pragma unroll
    for (int d = 0; d < DK; ++d) atomicAdd(op + d, w * mp[d]);
}

// ---------------------------------------------------------------------------
// LayerNorm over 128 features, per-type scale/bias.  One wave per node,
// 4 features per lane, wave32 shfl_xor reduction.
// ---------------------------------------------------------------------------
__global__ void ln_kernel(const float* __restrict__ pre,
                          const int* __restrict__ node_type,
                          const float* __restrict__ ln_scale,
                          const float* __restrict__ ln_bias,
                          float* __restrict__ out)
{
    const int wave = (blockIdx.x * blockDim.x + threadIdx.x) >> 5;
    const int lane = threadIdx.x & 31;
    if (wave >= NN) return;
    const float4 v = ((const float4*)(pre + (size_t)wave * HID))[lane];
    float s  = v.x + v.y + v.z + v.w;
    float sq = v.x * v.x + v.y * v.y + v.z * v.z + v.w * v.w;
    #pragma unroll
    for (int off = 16; off > 0; off >>= 1) {
        s  += __shfl_xor(s,  off, 32);
        sq += __shfl_xor(sq, off, 32);
    }
    const float mu  = s * (1.0f / HID);
    const float var = sq * (1.0f / HID) - mu * mu;
    const float inv = rsqrtf(var + 1e-5f);
    const int t = node_type[wave];
    const float4 sc = ((const float4*)(ln_scale + (size_t)t * HID))[lane];
    const float4 bi = ((const float4*)(ln_bias  + (size_t)t * HID))[lane];
    float4 o;
    o.x = (v.x - mu) * inv * sc.x + bi.x;
    o.y = (v.y - mu) * inv * sc.y + bi.y;
    o.z = (v.z - mu) * inv * sc.z + bi.z;
    o.w = (v.w - mu) * inv * sc.w + bi.w;
    ((float4*)(out + (size_t)wave * HID))[lane] = o;
}

// ---------------------------------------------------------------------------
extern "C" void kernel_launch(void* const* d_in, const int* in_sizes, int n_in,
                              void* d_out, int out_size, void* d_ws, size_t ws_size,
                              hipStream_t stream)
{
    const float* x         = (const float*)d_in[0];
    const int*   node_type = (const int*)  d_in[1];
    const int*   ei        = (const int*)  d_in[2];   // [2,E]
    const int*   erel      = (const int*)  d_in[3];
    const float* qW  = (const float*)d_in[4];
    const float* qb  = (const float*)d_in[5];
    const float* kW  = (const float*)d_in[6];
    const float* kb  = (const float*)d_in[7];
    const float* vW  = (const float*)d_in[8];
    const float* vb  = (const float*)d_in[9];
    const float* rel_att = (const float*)d_in[10];
    const float* rel_pri = (const float*)d_in[11];
    const float* rel_msg = (const float*)d_in[12];
    const float* upd_W = (const float*)d_in[13];
    const float* upd_b = (const float*)d_in[14];
    const float* ln_s  = (const float*)d_in[15];
    const float* ln_b  = (const float*)d_in[16];
    float* out = (float*)d_out;

    // workspace layout (floats)
    const size_t NHf = (size_t)NN * HID;
    float* ws    = (float*)d_ws;
    float* q     = ws;                       // N*128
    float* kq    = q    + NHf;               // N*128
    float* vv    = kq   + NHf;               // N*128
    float* k_rel = vv   + NHf;               // K_R*N*128
    float* v_rel = k_rel + (size_t)K_R * NHf;
    float* att   = v_rel + (size_t)K_R * NHf;        // E*8
    unsigned* mord = (unsigned*)(att + (size_t)EE * HEADS);  // N*8
    float* denom = (float*)(mord + (size_t)NN * HEADS);      // N*8
    float* aggr  = denom + (size_t)NN * HEADS;               // N*128
    float* pre   = aggr + NHf;                               // N*128

    const dim3 gTL(NN / 16, HID / 16);   // 3125 x 8, one wave per block
    typed_linear16<<<gTL, 32, 0, stream>>>(x, qW, qb, node_type, nullptr, q);
    typed_linear16<<<gTL, 32, 0, stream>>>(x, kW, kb, node_type, nullptr, kq);
    typed_linear16<<<gTL, 32, 0, stream>>>(x, vW, vb, node_type, nullptr, vv);

    const dim3 gRT(NN / 16, K_R * HEADS); // 3125 x 32
    rel_transform16<<<gRT, 32, 0, stream>>>(kq, vv, rel_att, rel_msg, k_rel, v_rel);

    init_kernel<<<(NN * HID + 255) / 256, 256, 0, stream>>>(mord, denom, aggr);

    const int edgeBlocks = (EE * HEADS + 255) / 256;
    score_kernel<<<edgeBlocks, 256, 0, stream>>>(q, k_rel, ei, erel, rel_pri, att, mord);
    exp_kernel  <<<edgeBlocks, 256, 0, stream>>>(ei, mord, att, denom);
    aggr_kernel <<<edgeBlocks, 256, 0, stream>>>(ei, erel, att, denom, v_rel, aggr);

    typed_linear16<<<gTL, 32, 0, stream>>>(aggr, upd_W, upd_b, node_type, x, pre);

    ln_kernel<<<(NN * 32 + 255) / 256, 256, 0, stream>>>(pre, node_type, ln_s, ln_b, out);
}